// GATv2_83700322664567
// MI455X (gfx1250) — compile-verified
//
#include <hip/hip_runtime.h>
#include <hip/hip_bf16.h>
#include <math.h>

#define HD      256
#define NHEAD   4
#define DHEAD   64
#define INF1    128
#define NEG_SLOPE 0.2f

typedef float v2f __attribute__((ext_vector_type(2)));
typedef float v8f __attribute__((ext_vector_type(8)));

// ---------------------------------------------------------------------------
// Order-preserving float <-> uint encoding for atomicMax-based segment max.
// encode is strictly monotonic; 0u is below encode(-inf), so 0u == "-inf".
// ---------------------------------------------------------------------------
__device__ __forceinline__ unsigned f2o(float f) {
    unsigned b = __float_as_uint(f);
    return (b & 0x80000000u) ? ~b : (b | 0x80000000u);
}
__device__ __forceinline__ float o2f(unsigned u) {
    unsigned b = (u & 0x80000000u) ? (u ^ 0x80000000u) : ~u;
    return __uint_as_float(b);
}

// ---------------------------------------------------------------------------
// Dual GEMM via V_WMMA_F32_16X16X4_F32:
//   outL = X @ Wl + bl ; outR = X @ Wr + br
// X: M x K row-major. W: K x HD row-major. One wave = one 16x16 tile of BOTH
// outputs (A fragment shared across the two WMMAs).
// A frag (ISA 7.12.2, f32 16x4): lanes 0-15 hold {K=k0,k0+1} of row M=lane,
// lanes 16-31 hold {K=k0+2,k0+3} of row M=lane-16.
// B frag mirrors A; C/D: VGPR i -> row (i | lane>=16 ? 8+i : i), col = lane&15.
// ---------------------------------------------------------------------------
__global__ void gemm_dual_wmma(const float* __restrict__ X,
                               const float* __restrict__ Wl, const float* __restrict__ bl,
                               const float* __restrict__ Wr, const float* __restrict__ br,
                               float* __restrict__ outL, float* __restrict__ outR,
                               int M, int K)
{
    const int lane = threadIdx.x & 31;
    const int wave = threadIdx.x >> 5;            // 0..3
    const int row0 = blockIdx.x * 16;             // M tile
    const int col0 = (blockIdx.y * 4 + wave) * 16; // N tile within HD
    const bool hi  = lane >= 16;
    const int  l15 = lane & 15;

    const float* __restrict__ xrow = X + (size_t)(row0 + l15) * K;
    const int bcol = col0 + l15;

    v8f accL = {};
    v8f accR = {};
    for (int k0 = 0; k0 < K; k0 += 4) {
        const int ka = k0 + (hi ? 2 : 0);
        v2f a;
        a.x = xrow[ka];
        a.y = xrow[ka + 1];
        v2f bL, bR;
        bL.x = Wl[(size_t)ka * HD + bcol];
        bL.y = Wl[(size_t)(ka + 1) * HD + bcol];
        bR.x = Wr[(size_t)ka * HD + bcol];
        bR.y = Wr[(size_t)(ka + 1) * HD + bcol];
        accL = __builtin_amdgcn_wmma_f32_16x16x4_f32(false, a, false, bL,
                                                     (short)0, accL, false, false);
        accR = __builtin_amdgcn_wmma_f32_16x16x4_f32(false, a, false, bR,
                                                     (short)0, accR, false, false);
    }

    const float bLv = bl[bcol];
    const float bRv = br[bcol];
#pragma unroll
    for (int i = 0; i < 8; ++i) {
        const int r = row0 + (hi ? 8 + i : i);
        outL[(size_t)r * HD + bcol] = accL[i] + bLv;
        outR[(size_t)r * HD + bcol] = accR[i] + bRv;
    }
}

// ---------------------------------------------------------------------------
// Per-edge attention scores: e[h] = sum_d leaky_relu(xl[src]+xr[dst]) * att[h]
// One wave per edge; lane covers d and d+32 per head. Wave-reduce, lane 0
// writes the score and does the encoded atomicMax segment-max.
// ---------------------------------------------------------------------------
__global__ void edge_scores(const float* __restrict__ xl, const float* __restrict__ xr,
                            const int* __restrict__ src, const int* __restrict__ dst,
                            const float* __restrict__ att,
                            float* __restrict__ escore, unsigned* __restrict__ emax,
                            int E)
{
    const int e = blockIdx.x * (blockDim.x >> 5) + (threadIdx.x >> 5);
    if (e >= E) return;
    const int lane = threadIdx.x & 31;
    const int s = src[e], d = dst[e];
    const float* __restrict__ pl = xl + (size_t)s * HD;
    const float* __restrict__ pr = xr + (size_t)d * HD;
#pragma unroll
    for (int h = 0; h < NHEAD; ++h) {
        float m0 = pl[h * DHEAD + lane]      + pr[h * DHEAD + lane];
        float m1 = pl[h * DHEAD + 32 + lane] + pr[h * DHEAD + 32 + lane];
        m0 = (m0 > 0.f) ? m0 : NEG_SLOPE * m0;
        m1 = (m1 > 0.f) ? m1 : NEG_SLOPE * m1;
        float v = m0 * att[h * DHEAD + lane] + m1 * att[h * DHEAD + 32 + lane];
#pragma unroll
        for (int off = 16; off > 0; off >>= 1)
            v += __shfl_xor(v, off, 32);
        if (lane == 0) {
            escore[(size_t)e * NHEAD + h] = v;
            atomicMax(&emax[(size_t)d * NHEAD + h], f2o(v));
        }
    }
}

// exp(e - max) and denominator accumulation; one thread per (edge, head).
__global__ void edge_exp(const int* __restrict__ dst,
                         float* __restrict__ escore, const unsigned* __restrict__ emax,
                         float* __restrict__ den, int E)
{
    const int i = blockIdx.x * blockDim.x + threadIdx.x;
    if (i >= E * NHEAD) return;
    const int e = i >> 2, h = i & 3;
    const int d = dst[e];
    const float mx = o2f(emax[(size_t)d * NHEAD + h]);
    const float ee = expf(escore[i] - mx);
    escore[i] = ee;
    atomicAdd(&den[(size_t)d * NHEAD + h], ee);
}

// out[dst] += alpha * xl[src]; one wave per edge, 8 f32 atomics per lane.
__global__ void edge_aggregate(const float* __restrict__ xl,
                               const int* __restrict__ src, const int* __restrict__ dst,
                               const float* __restrict__ escore, const float* __restrict__ den,
                               float* __restrict__ out, int E)
{
    const int e = blockIdx.x * (blockDim.x >> 5) + (threadIdx.x >> 5);
    if (e >= E) return;
    const int lane = threadIdx.x & 31;
    const int s = src[e], d = dst[e];
    const float* __restrict__ pl = xl + (size_t)s * HD;
    float* __restrict__ po = out + (size_t)d * HD;
    __builtin_prefetch(pl, 0, 1);   // global_prefetch_b8
#pragma unroll
    for (int h = 0; h < NHEAD; ++h) {
        const float alpha = escore[(size_t)e * NHEAD + h] /
                            (den[(size_t)d * NHEAD + h] + 1e-16f);
        atomicAdd(&po[h * DHEAD + lane],      alpha * pl[h * DHEAD + lane]);
        atomicAdd(&po[h * DHEAD + 32 + lane], alpha * pl[h * DHEAD + 32 + lane]);
    }
}

// per-(node,head) state reset
__global__ void zero_nodehead(unsigned* __restrict__ emax, float* __restrict__ den, int n)
{
    const int i = blockIdx.x * blockDim.x + threadIdx.x;
    if (i < n) { emax[i] = 0u; den[i] = 0.f; }
}

// out[n][c] = bias[c]
__global__ void init_bias(float* __restrict__ out, const float* __restrict__ bias, int total)
{
    const int i = blockIdx.x * blockDim.x + threadIdx.x;
    if (i < total) out[i] = bias[i & (HD - 1)];
}

__global__ void silu_inplace(float* __restrict__ h, int total)
{
    const int i = blockIdx.x * blockDim.x + threadIdx.x;
    if (i < total) {
        const float v = h[i];
        h[i] = v / (1.f + __expf(-v));
    }
}

__global__ void pool_zero(float* __restrict__ out, float* __restrict__ cnt, int total, int g)
{
    const int i = blockIdx.x * blockDim.x + threadIdx.x;
    if (i < total) out[i] = 0.f;
    if (i < g) cnt[i] = 0.f;
}

__global__ void pool_accum(const float* __restrict__ h, const int* __restrict__ batch,
                           float* __restrict__ out, float* __restrict__ cnt, int N)
{
    const int i = blockIdx.x * blockDim.x + threadIdx.x;
    const int n = i >> 8;
    const int c = i & (HD - 1);
    if (n >= N) return;
    const int g = batch[n];
    atomicAdd(&out[(size_t)g * HD + c], h[(size_t)n * HD + c]);
    if (c == 0) atomicAdd(&cnt[g], 1.0f);
}

__global__ void pool_final(float* __restrict__ out, const float* __restrict__ cnt, int total)
{
    const int i = blockIdx.x * blockDim.x + threadIdx.x;
    if (i < total) out[i] = out[i] / fmaxf(cnt[i >> 8], 1.0f);
}

// ---------------------------------------------------------------------------
extern "C" void kernel_launch(void* const* d_in, const int* in_sizes, int n_in,
                              void* d_out, int out_size, void* d_ws, size_t ws_size,
                              hipStream_t stream)
{
    const float* x     = (const float*)d_in[0];
    const int*   ei    = (const int*)  d_in[1];
    const int*   batch = (const int*)  d_in[2];
    const float* Wl1   = (const float*)d_in[3];
    const float* bl1   = (const float*)d_in[4];
    const float* Wr1   = (const float*)d_in[5];
    const float* br1   = (const float*)d_in[6];
    const float* att1  = (const float*)d_in[7];
    const float* bias1 = (const float*)d_in[8];
    const float* Wl2   = (const float*)d_in[9];
    const float* bl2   = (const float*)d_in[10];
    const float* Wr2   = (const float*)d_in[11];
    const float* br2   = (const float*)d_in[12];
    const float* att2  = (const float*)d_in[13];
    const float* bias2 = (const float*)d_in[14];
    (void)n_in; (void)ws_size;

    const int N = in_sizes[0] / INF1;      // 50000
    const int E = in_sizes[1] / 2;         // 800000
    const int G = out_size / HD;           // 64
    const int* src = ei;
    const int* dst = ei + E;

    // --- workspace carve-out --------------------------------------------
    char* ws = (char*)d_ws;
    size_t off = 0;
    auto carve = [&](size_t bytes) -> void* {
        void* p = ws + off;
        off += (bytes + 255) & ~(size_t)255;
        return p;
    };
    float*    A    = (float*)   carve((size_t)N * HD * sizeof(float)); // xl
    float*    B    = (float*)   carve((size_t)N * HD * sizeof(float)); // xr
    float*    C    = (float*)   carve((size_t)N * HD * sizeof(float)); // layer out
    float*    esc  = (float*)   carve((size_t)E * NHEAD * sizeof(float));
    unsigned* emax = (unsigned*)carve((size_t)N * NHEAD * sizeof(unsigned));
    float*    den  = (float*)   carve((size_t)N * NHEAD * sizeof(float));
    float*    cnt  = (float*)   carve((size_t)G * sizeof(float));
    float*    out  = (float*)d_out;

    const int NHDtot = N * HD;
    const int NHtot  = N * NHEAD;
    const int EHtot  = E * NHEAD;

    const dim3 gemmGrid((N + 15) / 16, HD / 64);
    const dim3 gemmBlk(128);
    const int  edgeBlks = (E + 7) / 8;             // 8 waves per 256-thread block

    // ================= Layer 1 =================
    gemm_dual_wmma<<<gemmGrid, gemmBlk, 0, stream>>>(x, Wl1, bl1, Wr1, br1, A, B, N, INF1);
    zero_nodehead<<<(NHtot + 255) / 256, 256, 0, stream>>>(emax, den, NHtot);
    edge_scores<<<edgeBlks, 256, 0, stream>>>(A, B, src, dst, att1, esc, emax, E);
    edge_exp<<<(EHtot + 255) / 256, 256, 0, stream>>>(dst, esc, emax, den, E);
    init_bias<<<(NHDtot + 255) / 256, 256, 0, stream>>>(C, bias1, NHDtot);
    edge_aggregate<<<edgeBlks, 256, 0, stream>>>(A, src, dst, esc, den, C, E);
    silu_inplace<<<(NHDtot + 255) / 256, 256, 0, stream>>>(C, NHDtot);

    // ================= Layer 2 =================
    gemm_dual_wmma<<<gemmGrid, gemmBlk, 0, stream>>>(C, Wl2, bl2, Wr2, br2, A, B, N, HD);
    zero_nodehead<<<(NHtot + 255) / 256, 256, 0, stream>>>(emax, den, NHtot);
    edge_scores<<<edgeBlks, 256, 0, stream>>>(A, B, src, dst, att2, esc, emax, E);
    edge_exp<<<(EHtot + 255) / 256, 256, 0, stream>>>(dst, esc, emax, den, E);
    init_bias<<<(NHDtot + 255) / 256, 256, 0, stream>>>(C, bias2, NHDtot);
    edge_aggregate<<<edgeBlks, 256, 0, stream>>>(A, src, dst, esc, den, C, E);

    // ================= Global mean pool =================
    pool_zero<<<(G * HD + 255) / 256, 256, 0, stream>>>(out, cnt, G * HD, G);
    pool_accum<<<(NHDtot + 255) / 256, 256, 0, stream>>>(C, batch, out, cnt, N);
    pool_final<<<(G * HD + 255) / 256, 256, 0, stream>>>(out, cnt, G * HD);
}